// RFFCGP_LAYER_39204461478737
// MI455X (gfx1250) — compile-verified
//
#include <hip/hip_runtime.h>
#include <math.h>

// ---------------------------------------------------------------------------
// RFF-CGP layer for MI455X (gfx1250), wave32, WMMA f32 16x16x4 + TDM staging.
// ---------------------------------------------------------------------------

typedef float v2f __attribute__((ext_vector_type(2)));
typedef float v8f __attribute__((ext_vector_type(8)));
typedef unsigned int u32x4 __attribute__((ext_vector_type(4)));
typedef int i32x4 __attribute__((ext_vector_type(4)));
typedef int i32x8 __attribute__((ext_vector_type(8)));

namespace {
constexpr int kB = 8, kN = 2048, kHD = 512, kH = 8, kM = 128, kVD = 64;
constexpr int kBH = kB * kH;   // 64
constexpr int kBN = kB * kN;   // 16384
constexpr int kKT = 64;        // K-chunk (rows of Phi staged per TDM transfer)
constexpr float kLAM = 0.01f;
constexpr float kEPS = 0.002f;
}

#define DEVI __device__ __forceinline__

extern __shared__ float g_smem[];

DEVI v8f wmma4(v2f a, v2f b, v8f c) {
  // V_WMMA_F32_16X16X4_F32 : D(16x16,f32) = A(16x4,f32) x B(4x16,f32) + C
  return __builtin_amdgcn_wmma_f32_16x16x4_f32(false, a, false, b, (short)0, c,
                                               false, false);
}

// element(i,k) = P[i*ld + k]   (row/i-major storage)
DEVI v2f ld_nm(const float* __restrict__ P, int i0, int k0, int ld, int lane) {
  int i = i0 + (lane & 15);
  int k = k0 + ((lane >> 4) << 1);
  const float* p = P + (size_t)i * ld + k;
  return v2f{p[0], p[1]};
}
// element(i,k) = P[k*ld + i]   (k-major storage)
DEVI v2f ld_km(const float* __restrict__ P, int i0, int k0, int ld, int lane) {
  int i = i0 + (lane & 15);
  int k = k0 + ((lane >> 4) << 1);
  const float* p = P + (size_t)k * ld + i;
  return v2f{p[0], p[ld]};
}

template <int NT>
DEVI void acc_zero(v8f (&a)[NT]) {
#pragma unroll
  for (int t = 0; t < NT; ++t)
#pragma unroll
    for (int r = 0; r < 8; ++r) a[t][r] = 0.0f;
}
template <int NT>
DEVI void acc_bias(v8f (&a)[NT], const float* __restrict__ bias, int col0,
                   int lane) {
#pragma unroll
  for (int t = 0; t < NT; ++t) {
    float bv = bias[col0 + t * 16 + (lane & 15)];
#pragma unroll
    for (int r = 0; r < 8; ++r) a[t][r] = bv;
  }
}

// Batched-load GEMM inner loop: stage CS k-steps of fragments in registers,
// then issue CS*NT WMMAs so loads overlap and waits amortize.
template <int NT, int CS, typename FA, typename FB>
DEVI void gemm_loop(v8f (&acc)[NT], int K, FA fa, FB fb) {
  for (int k0 = 0; k0 < K; k0 += 4 * CS) {
    v2f a4[CS];
    v2f b4[CS][NT];
#pragma unroll
    for (int s = 0; s < CS; ++s) {
      a4[s] = fa(k0 + 4 * s);
#pragma unroll
      for (int t = 0; t < NT; ++t) b4[s][t] = fb(k0 + 4 * s, t);
    }
#pragma unroll
    for (int s = 0; s < CS; ++s)
#pragma unroll
      for (int t = 0; t < NT; ++t) acc[t] = wmma4(a4[s], b4[s][t], acc[t]);
  }
}

// ---------------------------------------------------------------------------
// Tensor Data Mover: linear (1-D) tile copy global -> LDS.
// D# group0: count=1 | lds_addr | global_addr[56:0] | type=2
// D# group1: data_size=4B, tensor_dim0 = tile_dim0 = nelem (<= 65535)
// ---------------------------------------------------------------------------
DEVI void tdm_load_lin(const float* lds_dst, const float* gsrc,
                       unsigned nelem) {
  unsigned lds = (unsigned)(size_t)lds_dst;  // LDS byte offset (addr[31:0])
  unsigned long long ga = (unsigned long long)(size_t)gsrc;
  u32x4 g0;
  g0[0] = 1u;                               // count=1, user descriptor
  g0[1] = lds;                              // lds_addr (bytes)
  g0[2] = (unsigned)(ga & 0xFFFFFFFFu);     // global_addr[31:0]
  g0[3] = (unsigned)((ga >> 32) & 0x01FFFFFFu) | (2u << 30);  // addr[56:32]|type
  i32x8 g1;
  g1[0] = (int)(2u << 16);                  // data_size = 4 bytes
  g1[1] = (int)((nelem & 0xFFFFu) << 16);   // tensor_dim0[15:0]
  g1[2] = (int)((nelem >> 16) & 0xFFFFu);   // tensor_dim0[31:16]
  g1[3] = (int)((nelem & 0xFFFFu) << 16);   // tile_dim0
  g1[4] = 0; g1[5] = 0; g1[6] = 0; g1[7] = 0;
  i32x4 z4;
  z4[0] = 0; z4[1] = 0; z4[2] = 0; z4[3] = 0;
  i32x8 z8;
  z8[0] = 0; z8[1] = 0; z8[2] = 0; z8[3] = 0;
  z8[4] = 0; z8[5] = 0; z8[6] = 0; z8[7] = 0;
  __builtin_amdgcn_tensor_load_to_lds(g0, g1, z4, z4, z8, 0);
}

// ---------------------------------------------------------------------------
// K1: head projection  C = X @ W^T + b  -> head layout [(b*H+h)*N + n]*VD + d
// grid (BN/64, HD/64), block 128 (4 waves, each 16x64 tile)
// ---------------------------------------------------------------------------
__global__ void k_proj(const float* __restrict__ X, const float* __restrict__ W,
                       const float* __restrict__ bias,
                       const float* __restrict__ mask, float* __restrict__ Out) {
  const int lane = threadIdx.x & 31, wave = threadIdx.x >> 5;
  const int row0 = blockIdx.x * 64 + wave * 16;
  const int col0 = blockIdx.y * 64;  // == head * VD
  v8f acc[4];
  acc_bias(acc, bias, col0, lane);
  gemm_loop<4, 4>(
      acc, kHD, [&](int k0) { return ld_nm(X, row0, k0, kHD, lane); },
      [&](int k0, int t) { return ld_nm(W, col0 + t * 16, k0, kHD, lane); });
  const int nl = lane & 15, mo = (lane >> 4) * 8;
  const int h = blockIdx.y;  // VD == 64 == col tile
#pragma unroll
  for (int t = 0; t < 4; ++t) {
    int d = t * 16 + nl;
#pragma unroll
    for (int r = 0; r < 8; ++r) {
      int i = row0 + mo + r;
      int bb = i / kN, nn = i % kN;
      float v = acc[t][r];
      if (mask) v *= mask[bb * kN + nn];
      Out[((size_t)(bb * kH + h) * kN + nn) * kVD + d] = v;
    }
  }
}

// ---------------------------------------------------------------------------
// K2: RFF  Phi[bh][n][m] = 0.125*cos( Xh @ omega + b_rff[m] ) * mask[b][n]
// grid (BH, N/64, M/64), block 128
// ---------------------------------------------------------------------------
__global__ void k_rff(const float* __restrict__ Xh,
                      const float* __restrict__ omega,
                      const float* __restrict__ brff,
                      const float* __restrict__ mask,
                      float* __restrict__ Phi) {
  const int lane = threadIdx.x & 31, wave = threadIdx.x >> 5;
  const int bh = blockIdx.x;
  const int row0 = blockIdx.y * 64 + wave * 16;
  const int col0 = blockIdx.z * 64;
  const float* A = Xh + (size_t)bh * kN * kVD;
  v8f acc[4];
  acc_bias(acc, brff, col0, lane);
  gemm_loop<4, 4>(
      acc, kVD, [&](int k0) { return ld_nm(A, row0, k0, kVD, lane); },
      [&](int k0, int t) { return ld_km(omega, col0 + t * 16, k0, kM, lane); });
  const int nl = lane & 15, mo = (lane >> 4) * 8;
  const int bb = bh / kH;
  const float scale = 0.125f;  // sqrt(2/128)
  float* P = Phi + (size_t)bh * kN * kM;
#pragma unroll
  for (int t = 0; t < 4; ++t) {
    int m = col0 + t * 16 + nl;
#pragma unroll
    for (int r = 0; r < 8; ++r) {
      int n = row0 + mo + r;
      P[(size_t)n * kM + m] = scale * __cosf(acc[t][r]) * mask[bb * kN + n];
    }
  }
}

// ---------------------------------------------------------------------------
// K3: Gram  G[bh] = Phi^T Phi  (128x128, K = N = 2048)
// One block (256 thr, 8 waves) per bh; TDM double-buffers 64x128 Phi chunks
// (contiguous 32KB) into LDS; wave w computes rows [16w,16w+16) x all 128.
// dynamic LDS = 2 * kKT*kM floats = 64KB
// ---------------------------------------------------------------------------
__global__ void k_gram(const float* __restrict__ Phi, float* __restrict__ G) {
  const int lane = threadIdx.x & 31, wave = threadIdx.x >> 5;  // 8 waves
  const int bh = blockIdx.x;
  const int row0 = wave * 16;
  const float* P = Phi + (size_t)bh * kN * kM;
  v8f acc[8];
  acc_zero(acc);
  if (wave == 0) {
    tdm_load_lin(g_smem, P, kKT * kM);
    __builtin_amdgcn_s_wait_tensorcnt(0);
  }
  __syncthreads();
  const int nchunk = kN / kKT;  // 32
  for (int c = 0; c < nchunk; ++c) {
    const float* cur = g_smem + (c & 1) * (kKT * kM);
    if (wave == 0 && c + 1 < nchunk)
      tdm_load_lin(g_smem + ((c + 1) & 1) * (kKT * kM),
                   P + (size_t)(c + 1) * kKT * kM, kKT * kM);
    gemm_loop<8, 2>(
        acc, kKT, [&](int k0) { return ld_km(cur, row0, k0, kM, lane); },
        [&](int k0, int t) { return ld_km(cur, t * 16, k0, kM, lane); });
    if (wave == 0) __builtin_amdgcn_s_wait_tensorcnt(0);
    __syncthreads();
  }
  const int nl = lane & 15, mo = (lane >> 4) * 8;
  float* Gp = G + (size_t)bh * kM * kM;
#pragma unroll
  for (int t = 0; t < 8; ++t)
#pragma unroll
    for (int r = 0; r < 8; ++r)
      Gp[(size_t)(row0 + mo + r) * kM + t * 16 + nl] = acc[t][r];
}

// ---------------------------------------------------------------------------
// K4: Out[bh] (128 x VD) = Phi^T @ V   (K = N = 2048)
// One block (256 thr, 8 waves) per bh; TDM stages Phi (32KB) + V (16KB) chunks.
// dynamic LDS = (2*kKT*kM + 2*kKT*kVD) floats = 96KB
// ---------------------------------------------------------------------------
__global__ void k_phiT(const float* __restrict__ Phi, const float* __restrict__ V,
                       float* __restrict__ Out) {
  const int lane = threadIdx.x & 31, wave = threadIdx.x >> 5;  // 8 waves
  const int bh = blockIdx.x;
  const int row0 = wave * 16;
  const float* P = Phi + (size_t)bh * kN * kM;
  const float* Vp = V + (size_t)bh * kN * kVD;
  float* sP = g_smem;                    // 2 * kKT*kM
  float* sV = g_smem + 2 * kKT * kM;     // 2 * kKT*kVD
  v8f acc[4];
  acc_zero(acc);
  if (wave == 0) {
    tdm_load_lin(sP, P, kKT * kM);
    tdm_load_lin(sV, Vp, kKT * kVD);
    __builtin_amdgcn_s_wait_tensorcnt(0);
  }
  __syncthreads();
  const int nchunk = kN / kKT;  // 32
  for (int c = 0; c < nchunk; ++c) {
    const float* cP = sP + (c & 1) * (kKT * kM);
    const float* cV = sV + (c & 1) * (kKT * kVD);
    if (wave == 0 && c + 1 < nchunk) {
      tdm_load_lin(sP + ((c + 1) & 1) * (kKT * kM),
                   P + (size_t)(c + 1) * kKT * kM, kKT * kM);
      tdm_load_lin(sV + ((c + 1) & 1) * (kKT * kVD),
                   Vp + (size_t)(c + 1) * kKT * kVD, kKT * kVD);
    }
    gemm_loop<4, 4>(
        acc, kKT, [&](int k0) { return ld_km(cP, row0, k0, kM, lane); },
        [&](int k0, int t) { return ld_km(cV, t * 16, k0, kVD, lane); });
    if (wave == 0) __builtin_amdgcn_s_wait_tensorcnt(0);
    __syncthreads();
  }
  const int nl = lane & 15, mo = (lane >> 4) * 8;
  float* O = Out + (size_t)bh * kM * kVD;
#pragma unroll
  for (int t = 0; t < 4; ++t)
#pragma unroll
    for (int r = 0; r < 8; ++r)
      O[(size_t)(row0 + mo + r) * kVD + t * 16 + nl] = acc[t][r];
}

// ---------------------------------------------------------------------------
// K5: C[bh] (N x VD) = Phi @ Wm ; mode 0: store C ; mode 1: store (V - C)/LAM
// grid (BH, N/64), block 128
// ---------------------------------------------------------------------------
__global__ void k_phimat(const float* __restrict__ Phi,
                         const float* __restrict__ Wm,
                         const float* __restrict__ V, float* __restrict__ Out,
                         int mode) {
  const int lane = threadIdx.x & 31, wave = threadIdx.x >> 5;
  const int bh = blockIdx.x;
  const int row0 = blockIdx.y * 64 + wave * 16;
  const float* P = Phi + (size_t)bh * kN * kM;
  const float* Wp = Wm + (size_t)bh * kM * kVD;
  v8f acc[4];
  acc_zero(acc);
  gemm_loop<4, 4>(
      acc, kM, [&](int k0) { return ld_nm(P, row0, k0, kM, lane); },
      [&](int k0, int t) { return ld_km(Wp, t * 16, k0, kVD, lane); });
  const int nl = lane & 15, mo = (lane >> 4) * 8;
  const float ilam = 1.0f / kLAM;
  float* O = Out + (size_t)bh * kN * kVD;
  const float* Vp = V ? V + (size_t)bh * kN * kVD : nullptr;
#pragma unroll
  for (int t = 0; t < 4; ++t)
#pragma unroll
    for (int r = 0; r < 8; ++r) {
      size_t idx = (size_t)(row0 + mo + r) * kVD + t * 16 + nl;
      float v = acc[t][r];
      if (mode == 1) v = (Vp[idx] - v) * ilam;
      O[idx] = v;
    }
}

// ---------------------------------------------------------------------------
// K6: batched small matmul  C[bh](128 x nc) = A[bh](128x128) @ Bm[bh](128 x nc)
// grid (BH, 2, nc/64), block 128
// ---------------------------------------------------------------------------
__global__ void k_bmm(const float* __restrict__ A, const float* __restrict__ Bm,
                      float* __restrict__ C, int nc) {
  const int lane = threadIdx.x & 31, wave = threadIdx.x >> 5;
  const int bh = blockIdx.x;
  const int row0 = blockIdx.y * 64 + wave * 16;
  const int col0 = blockIdx.z * 64;
  const float* Ap = A + (size_t)bh * kM * kM;
  const float* Bp = Bm + (size_t)bh * kM * nc;
  v8f acc[4];
  acc_zero(acc);
  gemm_loop<4, 4>(
      acc, kM, [&](int k0) { return ld_nm(Ap, row0, k0, kM, lane); },
      [&](int k0, int t) { return ld_km(Bp, col0 + t * 16, k0, nc, lane); });
  const int nl = lane & 15, mo = (lane >> 4) * 8;
  float* Cp = C + (size_t)bh * kM * nc;
#pragma unroll
  for (int t = 0; t < 4; ++t)
#pragma unroll
    for (int r = 0; r < 8; ++r)
      Cp[(size_t)(row0 + mo + r) * nc + col0 + t * 16 + nl] = acc[t][r];
}

// ---------------------------------------------------------------------------
// K7: output GEMM  Out[i][j] = sum_k Yhead(i,k) * Wo[j][k] + bo[j]
// grid (BN/64, HD/64), block 128
// ---------------------------------------------------------------------------
__global__ void k_outgemm(const float* __restrict__ Y,
                          const float* __restrict__ Wo,
                          const float* __restrict__ bo,
                          float* __restrict__ Out) {
  const int lane = threadIdx.x & 31, wave = threadIdx.x >> 5;
  const int row0 = blockIdx.x * 64 + wave * 16;
  const int col0 = blockIdx.y * 64;
  v8f acc[4];
  acc_bias(acc, bo, col0, lane);
  const int i = row0 + (lane & 15);
  const int bb = i / kN, nn = i % kN;
  gemm_loop<4, 4>(
      acc, kHD,
      [&](int k0) {
        int k = k0 + ((lane >> 4) << 1);
        int h = k >> 6, d = k & 63;  // pair (k,k+1) stays inside one head
        const float* pa = Y + ((size_t)(bb * kH + h) * kN + nn) * kVD + d;
        return v2f{pa[0], pa[1]};
      },
      [&](int k0, int t) { return ld_nm(Wo, col0 + t * 16, k0, kHD, lane); });
  const int nl = lane & 15, mo = (lane >> 4) * 8;
#pragma unroll
  for (int t = 0; t < 4; ++t)
#pragma unroll
    for (int r = 0; r < 8; ++r)
      Out[(size_t)(row0 + mo + r) * kHD + col0 + t * 16 + nl] = acc[t][r];
}

// ---------------------------------------------------------------------------
// K8: batched Cholesky + two triangular solves (LDS resident).
// A = s1*G1 + s2*G2 + dadd*I ; chol(A)=L ; solve L L^T X = RHS
// RHS = identity if Rhs==nullptr, else Rhs[bh] (128 x nc).
// grid (BH), block 128, dynamic LDS = (16384 + 128*nc)*4 bytes
// ---------------------------------------------------------------------------
__global__ void k_chol(const float* __restrict__ G1, float s1,
                       const float* __restrict__ G2, float s2, float dadd,
                       const float* __restrict__ Rhs, int nc,
                       float* __restrict__ Xout, float* __restrict__ logdet) {
  float* sA = g_smem;            // 128*128
  float* sY = g_smem + kM * kM;  // 128*nc
  const int bh = blockIdx.x, tid = threadIdx.x;
  const size_t ob = (size_t)bh * kM * kM;
  for (int idx = tid; idx < kM * kM; idx += 128) {
    float v = s1 * G1[ob + idx];
    if (G2) v += s2 * G2[ob + idx];
    sA[idx] = v;
  }
  __syncthreads();
  sA[tid * kM + tid] += dadd;  // tid in [0,128)
  __syncthreads();
  // Cholesky, thread i owns row i (lower triangle)
  for (int k = 0; k < kM; ++k) {
    if (tid == k) sA[k * kM + k] = sqrtf(sA[k * kM + k]);
    __syncthreads();
    float lkk = sA[k * kM + k];
    if (tid > k) sA[tid * kM + k] /= lkk;
    __syncthreads();
    if (tid > k) {
      float lik = sA[tid * kM + k];
      for (int j = k + 1; j <= tid; ++j) sA[tid * kM + j] -= lik * sA[j * kM + k];
    }
    __syncthreads();
  }
  if (logdet && tid == 0) {
    float s = 0.0f;
    for (int k = 0; k < kM; ++k) s += logf(sA[k * kM + k]);
    logdet[bh] = 2.0f * s;
  }
  // column-per-thread forward then in-place backward solve
  const int j = tid;
  if (j < nc) {
    for (int r = 0; r < kM; ++r) {
      float v = Rhs ? Rhs[(size_t)bh * kM * nc + (size_t)r * nc + j]
                    : (r == j ? 1.0f : 0.0f);
      for (int t = 0; t < r; ++t) v -= sA[r * kM + t] * sY[t * nc + j];
      sY[r * nc + j] = v / sA[r * kM + r];
    }
    for (int r = kM - 1; r >= 0; --r) {
      float v = sY[r * nc + j];
      for (int t = r + 1; t < kM; ++t) v -= sA[t * kM + r] * sY[t * nc + j];
      sY[r * nc + j] = v / sA[r * kM + r];
    }
  }
  __syncthreads();
  for (int idx = tid; idx < kM * nc; idx += 128)
    Xout[(size_t)bh * kM * nc + idx] = sY[idx];
}

// ---------------------------------------------------------------------------
// Reductions
// ---------------------------------------------------------------------------
DEVI float block_reduce(float v, float* red) {
  const int tid = threadIdx.x;
  red[tid] = v;
  __syncthreads();
  for (int s = blockDim.x >> 1; s > 0; s >>= 1) {
    if (tid < s) red[tid] += red[tid + s];
    __syncthreads();
  }
  float r = red[0];
  __syncthreads();
  return r;
}

__global__ void k_vv(const float* __restrict__ V, float* __restrict__ yy) {
  __shared__ float red[256];
  const int bh = blockIdx.x;
  float s = 0.0f;
  const float* Vp = V + (size_t)bh * kN * kVD;
  for (int i = threadIdx.x; i < kN * kVD; i += blockDim.x) {
    float v = Vp[i];
    s += v * v;
  }
  s = block_reduce(s, red);
  if (threadIdx.x == 0) yy[bh] = s;
}

__global__ void k_reg(const float* __restrict__ GAinv,
                      const float* __restrict__ Cq, const float* __restrict__ Pq,
                      const float* __restrict__ Ck, const float* __restrict__ Pk,
                      const float* __restrict__ Tk,
                      const float* __restrict__ PhiTy,
                      const float* __restrict__ SqY,
                      const float* __restrict__ yy,
                      const float* __restrict__ ldAo,
                      const float* __restrict__ ldSq,
                      const float* __restrict__ ldSk,
                      float* __restrict__ reg_out) {
  __shared__ float red[256];
  const int bh = blockIdx.x, tid = threadIdx.x;
  const size_t o2 = (size_t)bh * kM * kM, o3 = (size_t)bh * kM * kVD;
  float sGCq = 0, sGPq = 0, sGCk = 0, sGPk = 0, sTkCk = 0, trCk = 0, sPhi = 0;
  for (int i = tid; i < kM * kM; i += 256) {
    float g = GAinv[o2 + i];
    sGCq += g * Cq[o2 + i];
    sGPq += g * Pq[o2 + i];
    sGCk += g * Ck[o2 + i];
    sGPk += g * Pk[o2 + i];
    sTkCk += Tk[o2 + i] * Ck[o2 + i];
    if (i % (kM + 1) == 0) trCk += Ck[o2 + i];
  }
  for (int i = tid; i < kM * kVD; i += 256) sPhi += PhiTy[o3 + i] * SqY[o3 + i];
  sGCq = block_reduce(sGCq, red);
  sGPq = block_reduce(sGPq, red);
  sGCk = block_reduce(sGCk, red);
  sGPk = block_reduce(sGPk, red);
  sTkCk = block_reduce(sTkCk, red);
  trCk = block_reduce(trCk, red);
  sPhi = block_reduce(sPhi, red);
  if (tid == 0) {
    const float D = (float)kVD;
    const float iE = 1.0f / kEPS, iE2 = iE * iE;
    float model_q = D * (sGCq * iE - sGPq * iE2);
    float model_k = D * (sGCk * iE - sGPk * iE2);
    float datafit_q = yy[bh] * iE - sPhi * iE2;
    float datafit_k = D * (trCk * iE - sTkCk * iE2);
    // logdet(I + (LAM/EPS) Ao^-1 C) = M*log(LAM) + logdet(S) - logdet(Ao)
    float logc = (float)kN * logf(kEPS) + (float)kM * logf(kLAM);
    float logdet_q = logc + ldSq[bh] - ldAo[bh];
    float logdet_k = logc + ldSk[bh] - ldAo[bh];
    float cst = (float)kN * logf(6.283185307179586f);
    float LBq = -0.5f * (datafit_q + model_q + logdet_q + cst);
    float LBk = -0.5f * (datafit_k + model_k + logdet_k + cst);
    atomicAdd(reg_out, -(LBq + LBk) / (float)kBH);
  }
}

// ---------------------------------------------------------------------------
// Host orchestration
// ---------------------------------------------------------------------------
extern "C" void kernel_launch(void* const* d_in, const int* in_sizes, int n_in,
                              void* d_out, int out_size, void* d_ws,
                              size_t ws_size, hipStream_t stream) {
  (void)in_sizes; (void)n_in; (void)out_size; (void)ws_size;
  const float* x    = (const float*)d_in[0];
  const float* mask = (const float*)d_in[1];
  const float* Wq = (const float*)d_in[2];  const float* bq  = (const float*)d_in[3];
  const float* Wk = (const float*)d_in[4];  const float* bk  = (const float*)d_in[5];
  const float* Wv = (const float*)d_in[6];  const float* bv  = (const float*)d_in[7];
  const float* Wx0= (const float*)d_in[8];  const float* bx0 = (const float*)d_in[9];
  const float* Wo = (const float*)d_in[10]; const float* bo  = (const float*)d_in[11];
  const float* omega = (const float*)d_in[12];
  const float* brff  = (const float*)d_in[13];
  float* out = (float*)d_out;

  // workspace carve-up (floats)
  float* w = (float*)d_ws;
  size_t off = 0;
  auto alloc = [&](size_t n) { float* p = w + off; off += n; return p; };
  const size_t szHead = (size_t)kBH * kN * kVD;   // 8.39M
  const size_t szPhi  = (size_t)kBH * kN * kM;    // 16.8M
  const size_t szMM   = (size_t)kBH * kM * kM;    // 1.05M
  const size_t szMD   = (size_t)kBH * kM * kVD;   // 0.52M
  float* tmp  = alloc(szHead);   // head projections, later y
  float* v    = alloc(szHead);
  float* u    = alloc(szHead);
  float* PhiA = alloc(szPhi);    // Phi_o then Phi_q
  float* PhiB = alloc(szPhi);    // Phi_k
  float* Go   = alloc(szMM);
  float* Cq   = alloc(szMM);
  float* Gk   = alloc(szMM);
  float* AoInv= alloc(szMM);
  float* GAinv= alloc(szMM);     // == B_o
  float* SqInv= alloc(szMM);
  float* SkInv= alloc(szMM);
  float* Tq   = alloc(szMM);
  float* Pq   = alloc(szMM);
  float* Tk   = alloc(szMM);
  float* Pk   = alloc(szMM);
  float* PkTv = alloc(szMD);
  float* AkPv = alloc(szMD);
  float* PkTu = alloc(szMD);
  float* mid  = alloc(szMD);
  float* PhiTy= alloc(szMD);
  float* SqY  = alloc(szMD);
  float* yy   = alloc(kBH);
  float* ldAo = alloc(kBH);
  float* ldSq = alloc(kBH);
  float* ldSk = alloc(kBH);

  const dim3 blk(128);
  const dim3 blk256(256);
  const dim3 gProj(kBN / 64, kHD / 64);
  const dim3 gRff(kBH, kN / 64, kM / 64);
  const dim3 gPhiMat(kBH, kN / 64);
  const size_t lsGram = (size_t)2 * kKT * kM * sizeof(float);              // 64KB
  const size_t lsPhiT = ((size_t)2 * kKT * kM + 2 * kKT * kVD) * sizeof(float);  // 96KB
  const size_t ls128 = (size_t)(kM * kM + kM * kM) * sizeof(float);        // 128KB
  const size_t ls64  = (size_t)(kM * kM + kM * kVD) * sizeof(float);       // 96KB

  // projections + features + grams
  k_proj<<<gProj, blk, 0, stream>>>(x, Wv, bv, mask, v);
  k_proj<<<gProj, blk, 0, stream>>>(x, Wx0, bx0, nullptr, tmp);
  k_rff<<<gRff, blk, 0, stream>>>(tmp, omega, brff, mask, PhiA);   // Phi_o
  k_gram<<<kBH, blk256, lsGram, stream>>>(PhiA, Go);
  k_proj<<<gProj, blk, 0, stream>>>(x, Wq, bq, nullptr, tmp);
  k_rff<<<gRff, blk, 0, stream>>>(tmp, omega, brff, mask, PhiA);   // Phi_q
  k_gram<<<kBH, blk256, lsGram, stream>>>(PhiA, Cq);
  k_proj<<<gProj, blk, 0, stream>>>(x, Wk, bk, nullptr, tmp);
  k_rff<<<gRff, blk, 0, stream>>>(tmp, omega, brff, mask, PhiB);   // Phi_k
  k_gram<<<kBH, blk256, lsGram, stream>>>(PhiB, Gk);

  // u = (v - Phi_k Ak^-1 Phi_k^T v)/LAM
  k_phiT<<<kBH, blk256, lsPhiT, stream>>>(PhiB, v, PkTv);
  k_chol<<<kBH, blk, ls64, stream>>>(Gk, 1.0f, nullptr, 0.0f, kLAM,
                                     PkTv, kVD, AkPv, nullptr);
  k_phimat<<<gPhiMat, blk, 0, stream>>>(PhiB, AkPv, v, u, 1);
  k_phiT<<<kBH, blk256, lsPhiT, stream>>>(PhiB, u, PkTu);

  // Ao^-1, B_o = Go Ao^-1, mid, y, out
  k_chol<<<kBH, blk, ls128, stream>>>(Go, 1.0f, nullptr, 0.0f, kLAM,
                                      nullptr, kM, AoInv, ldAo);
  k_bmm<<<dim3(kBH, 2, 2), blk, 0, stream>>>(Go, AoInv, GAinv, kM);
  k_bmm<<<dim3(kBH, 2, 1), blk, 0, stream>>>(GAinv, PkTu, mid, kVD);
  k_phimat<<<gPhiMat, blk, 0, stream>>>(PhiA, mid, nullptr, tmp, 0);  // y
  k_outgemm<<<gProj, blk, 0, stream>>>(tmp, Wo, bo, out);

  // regularizer pieces
  k_chol<<<kBH, blk, ls128, stream>>>(Go, 1.0f / kLAM, Cq, 1.0f / kEPS, 1.0f,
                                      nullptr, kM, SqInv, ldSq);
  k_chol<<<kBH, blk, ls128, stream>>>(Go, 1.0f / kLAM, Gk, 1.0f / kEPS, 1.0f,
                                      nullptr, kM, SkInv, ldSk);
  k_phiT<<<kBH, blk256, lsPhiT, stream>>>(PhiA, v, PhiTy);
  k_bmm<<<dim3(kBH, 2, 1), blk, 0, stream>>>(SqInv, PhiTy, SqY, kVD);
  k_bmm<<<dim3(kBH, 2, 2), blk, 0, stream>>>(SqInv, Cq, Tq, kM);
  k_bmm<<<dim3(kBH, 2, 2), blk, 0, stream>>>(Cq, Tq, Pq, kM);
  k_bmm<<<dim3(kBH, 2, 2), blk, 0, stream>>>(SkInv, Gk, Tk, kM);
  k_bmm<<<dim3(kBH, 2, 2), blk, 0, stream>>>(Gk, Tk, Pk, kM);
  k_vv<<<kBH, 256, 0, stream>>>(v, yy);

  (void)hipMemsetAsync(out + (size_t)kBN * kHD, 0, sizeof(float), stream);
  k_reg<<<kBH, 256, 0, stream>>>(GAinv, Cq, Pq, Gk, Pk, Tk, PhiTy, SqY, yy,
                                 ldAo, ldSq, ldSk, out + (size_t)kBN * kHD);
}